// Shift_gcn_49709951484724
// MI455X (gfx1250) — compile-verified
//
#include <hip/hip_runtime.h>
#include <hip/hip_bf16.h>
#include <math.h>

#define B_ 32
#define T_ 512
#define V_ 25
#define C_ 64
#define D_ 128
#define N_ (B_ * T_)          // 16384
#define NR_ (N_ * V_)         // 409600 rows for residual GEMM
#define EPS_ 1e-5f
#define CP_ (C_ + 1)          // padded LDS row (bank-conflict-free A reads)

typedef float v2f __attribute__((ext_vector_type(2)));
typedef float v8f __attribute__((ext_vector_type(8)));

static __device__ __forceinline__ v8f wmma_f32(v2f a, v2f b, v8f c) {
    // D = A(16x4 f32) * B(4x16 f32) + C(16x16 f32)
    return __builtin_amdgcn_wmma_f32_16x16x4_f32(
        /*neg_a=*/false, a, /*neg_b=*/false, b,
        /*c_mod=*/(short)0, c, /*reuse_a=*/false, /*reuse_b=*/false);
}

// ---------------------------------------------------------------------------
// Pass 0: precompute tanh(mask)+1 and inverse of shift_out permutation
// ---------------------------------------------------------------------------
__global__ __launch_bounds__(256) void init_kernel(
    const float* __restrict__ fmask, const int* __restrict__ shift_out,
    float* __restrict__ maskf, int* __restrict__ inv_out)
{
    int tid = threadIdx.x;
    for (int i = tid; i < V_ * C_; i += 256)
        maskf[i] = tanhf(fmask[i]) + 1.0f;
    for (int f = tid; f < V_ * D_; f += 256)
        inv_out[shift_out[f]] = f;
}

// ---------------------------------------------------------------------------
// Pass 1: y = shift_out( (shift_in(x0) * mask) @ W + lin_b )
// One block = 8 waves = all 8 d-tiles for one v; loops over 8 n-tiles,
// staging the gathered+masked A tile in LDS once (8x less gather traffic).
// Writes directly in (B, D, T, V) layout; accumulates BN1d stats.
// grid = (128 n-supertiles, 25 v)
// ---------------------------------------------------------------------------
__global__ __launch_bounds__(256) void gemm1_kernel(
    const float* __restrict__ x0,       // (N_, V_*C_)
    const float* __restrict__ W,        // (C_, D_)
    const float* __restrict__ lin_b,    // (D_)
    const int*   __restrict__ shift_in, // (V_*C_)
    const float* __restrict__ maskf,    // (V_*C_)
    const int*   __restrict__ inv_out,  // (V_*D_)
    float* __restrict__ out,            // (B_, D_, T_, V_)
    float* __restrict__ ysum,           // (V_*D_)
    float* __restrict__ ysumsq)         // (V_*D_)
{
    __shared__ int   s_sidx[C_];
    __shared__ float s_mask[C_];
    __shared__ float s_w[C_ * D_];      // 32 KB: full W, staged once
    __shared__ float s_a[16 * CP_];     // padded gathered+masked A tile

    const int tid = threadIdx.x;
    const int v   = blockIdx.y;

    if (tid < C_) {
        s_sidx[tid] = shift_in[v * C_ + tid];
        s_mask[tid] = maskf[v * C_ + tid];
    }
    for (int i = tid; i < C_ * D_; i += 256)
        s_w[i] = W[i];                  // same row-major (C,D) layout

    const int wave = tid >> 5;
    const int lane = tid & 31;
    const int half = lane >> 4;
    const int ml   = lane & 15;

    const int   d   = wave * 16 + ml;   // this lane's output column
    const float lb  = lin_b[d];
    const int   dst = inv_out[v * D_ + d];
    const int   vp  = dst >> 7;         // dst / D_
    const int   dp  = dst & 127;        // dst % D_

    float cs = 0.0f, cq = 0.0f;

    for (int it = 0; it < 8; ++it) {
        const int n0 = (blockIdx.x * 8 + it) * 16;
        __syncthreads();                // previous tile fully consumed
        #pragma unroll
        for (int j = 0; j < 4; ++j) {   // stage 16x64 gathered A (4/thread)
            const int idx = tid + 256 * j;
            const int row = idx >> 6, c = idx & 63;
            s_a[row * CP_ + c] =
                x0[(size_t)(n0 + row) * (V_ * C_) + s_sidx[c]] * s_mask[c];
        }
        __syncthreads();

        v8f acc = {};
        #pragma unroll
        for (int k0 = 0; k0 < C_; k0 += 4) {
            const int k = k0 + half * 2;
            v2f a, b;
            a.x = s_a[ml * CP_ + k];
            a.y = s_a[ml * CP_ + k + 1];
            b.x = s_w[k * D_ + d];
            b.y = s_w[(k + 1) * D_ + d];
            acc = wmma_f32(a, b, acc);
        }

        const int bb = n0 >> 9;         // b constant within aligned tile
        #pragma unroll
        for (int i = 0; i < 8; ++i) {
            const int n = n0 + i + half * 8;
            const float y = acc[i] + lb;
            cs += y;
            cq += y * y;
            const int t = n & 511;
            out[(((size_t)bb * D_ + dp) * T_ + t) * V_ + vp] = y;
        }
    }
    // combine lane L (rows 0..7) with lane L^16 (rows 8..15)
    cs += __shfl_xor(cs, 16, 32);
    cq += __shfl_xor(cq, 16, 32);
    if (half == 0) {
        atomicAdd(&ysum[dst], cs);
        atomicAdd(&ysumsq[dst], cq);
    }
}

// ---------------------------------------------------------------------------
// Pass 2: second-moment S2 = X^T X (64x64) and column sums S1 of x0 rows,
// via WMMA rank-k update. grid = 400 blocks x 512 threads (16 waves = 4x4 tiles)
// ---------------------------------------------------------------------------
__global__ __launch_bounds__(512) void moment_kernel(
    const float* __restrict__ x0,   // (NR_, C_)
    float* __restrict__ S1,         // (C_)
    float* __restrict__ S2)         // (C_, C_)
{
    __shared__ float s_x[64 * C_];  // 64 rows staged per iteration (16 KB)

    const int tid  = threadIdx.x;
    const int wave = tid >> 5;
    const int lane = tid & 31;
    const int half = lane >> 4;
    const int ml   = lane & 15;
    const int ti   = (wave >> 2) * 16;
    const int tj   = (wave & 3) * 16;

    v8f acc = {};
    float s1p = 0.0f;
    const size_t base = (size_t)blockIdx.x * 1024 * C_;

    for (int it = 0; it < 16; ++it) {
        __syncthreads();
        #pragma unroll
        for (int j = 0; j < 8; ++j) {
            const int id = tid + 512 * j;              // coalesced
            const float vv = x0[base + (size_t)it * 4096 + id];
            s_x[id] = vv;
            s1p += vv;                                 // col = tid % 64, fixed
        }
        if (it < 15)                                   // global_prefetch_b8
            __builtin_prefetch(&x0[base + (size_t)(it + 1) * 4096 + tid * 8], 0, 0);
        __syncthreads();
        #pragma unroll
        for (int k0 = 0; k0 < C_; k0 += 4) {
            const int k = k0 + half * 2;
            v2f a, b;
            a.x = s_x[k * C_ + ti + ml];               // A = X^T
            a.y = s_x[(k + 1) * C_ + ti + ml];
            b.x = s_x[k * C_ + tj + ml];               // B = X
            b.y = s_x[(k + 1) * C_ + tj + ml];
            acc = wmma_f32(a, b, acc);
        }
    }
    atomicAdd(&S1[tid & 63], s1p);
    #pragma unroll
    for (int i = 0; i < 8; ++i) {
        const int m  = ti + i + half * 8;
        const int nn = tj + ml;
        atomicAdd(&S2[m * C_ + nn], acc[i]);
    }
}

// ---------------------------------------------------------------------------
// Pass 3: fold both BN affines into per-feature scale/shift
// ---------------------------------------------------------------------------
__global__ __launch_bounds__(256) void finalize_kernel(
    const float* __restrict__ ysum, const float* __restrict__ ysumsq,
    const float* __restrict__ gamma, const float* __restrict__ beta,
    const float* __restrict__ S1, const float* __restrict__ S2,
    const float* __restrict__ down_w, const float* __restrict__ down_b,
    const float* __restrict__ dg, const float* __restrict__ db,
    float* __restrict__ a_arr, float* __restrict__ b2_arr,
    float* __restrict__ ra, float* __restrict__ rb)
{
    const int tid = threadIdx.x;
    const float invN = 1.0f / (float)N_;
    for (int f = tid; f < V_ * D_; f += 256) {
        const float mu  = ysum[f] * invN;
        const float var = ysumsq[f] * invN - mu * mu;
        const float sc  = gamma[f] * rsqrtf(var + EPS_);
        a_arr[f]  = sc;
        b2_arr[f] = beta[f] - mu * sc;
    }
    if (tid < D_) {
        const float* w = down_w + tid * C_;
        const float invR = 1.0f / (float)NR_;
        float m = 0.0f;
        for (int c = 0; c < C_; ++c) m += w[c] * S1[c];
        m *= invR;                                   // E[w.x]
        float q = 0.0f;
        for (int c1 = 0; c1 < C_; ++c1) {
            float acc = 0.0f;
            for (int c2 = 0; c2 < C_; ++c2) acc += S2[c1 * C_ + c2] * w[c2];
            q += w[c1] * acc;
        }
        q *= invR;                                   // E[(w.x)^2]
        const float rvar = q - m * m;
        const float sc = dg[tid] * rsqrtf(rvar + EPS_);
        ra[tid] = sc;
        rb[tid] = db[tid] - sc * m;                  // down_b folds out of rb
    }
}

// ---------------------------------------------------------------------------
// Pass 4: residual GEMM (WMMA) + dual BN affine + ReLU, in-place on d_out.
// grid = 25600 row-tiles, 8 waves = 8 d-tiles each
// ---------------------------------------------------------------------------
__global__ __launch_bounds__(256) void gemm2_fuse_kernel(
    const float* __restrict__ x0,      // (NR_, C_)
    const float* __restrict__ down_w,  // (D_, C_)
    const float* __restrict__ a_arr, const float* __restrict__ b2_arr,
    const float* __restrict__ ra, const float* __restrict__ rb,
    float* __restrict__ out)           // (B_, D_, T_, V_)
{
    __shared__ float s_a[16 * CP_];    // padded 16 rows of x0, shared by 8 waves

    const int tid  = threadIdx.x;
    const int wave = tid >> 5;
    const int lane = tid & 31;
    const int half = lane >> 4;
    const int ml   = lane & 15;
    const int r0   = blockIdx.x * 16;

    for (int i = tid; i < 16 * C_; i += 256) {
        const int row = i >> 6, c = i & 63;
        s_a[row * CP_ + c] = x0[(size_t)r0 * C_ + i];   // coalesced
    }
    __syncthreads();

    const int d = wave * 16 + ml;
    const float* wr = down_w + d * C_;

    v8f acc = {};
    #pragma unroll
    for (int k0 = 0; k0 < C_; k0 += 4) {
        const int k = k0 + half * 2;
        v2f a, b;
        a.x = s_a[ml * CP_ + k];
        a.y = s_a[ml * CP_ + k + 1];
        b.x = wr[k];
        b.y = wr[k + 1];
        acc = wmma_f32(a, b, acc);
    }

    const float rad = ra[d], rbd = rb[d];
    const int bb = r0 / (T_ * V_);     // constant within aligned 16-row tile
    #pragma unroll
    for (int i = 0; i < 8; ++i) {
        const int r   = r0 + i + half * 8;
        const int rem = r - bb * T_ * V_;
        const int t   = rem / V_;
        const int v   = rem - t * V_;
        const int f   = v * D_ + d;
        const size_t addr = (((size_t)bb * D_ + d) * T_ + t) * V_ + v;
        const float y   = out[addr];
        const float val = a_arr[f] * y + b2_arr[f] + rad * acc[i] + rbd;
        out[addr] = fmaxf(val, 0.0f);
    }
}

// ---------------------------------------------------------------------------
extern "C" void kernel_launch(void* const* d_in, const int* in_sizes, int n_in,
                              void* d_out, int out_size, void* d_ws, size_t ws_size,
                              hipStream_t stream) {
    const float* x0        = (const float*)d_in[0];
    const float* fmask     = (const float*)d_in[1];
    const float* W         = (const float*)d_in[2];
    const float* lin_b     = (const float*)d_in[3];
    const float* bn_gamma  = (const float*)d_in[4];
    const float* bn_beta   = (const float*)d_in[5];
    const float* down_w    = (const float*)d_in[6];
    const float* down_b    = (const float*)d_in[7];
    const float* dbn_gamma = (const float*)d_in[8];
    const float* dbn_beta  = (const float*)d_in[9];
    const int*   shift_in  = (const int*)d_in[10];
    const int*   shift_out = (const int*)d_in[11];
    float* out = (float*)d_out;
    char*  ws  = (char*)d_ws;

    float* maskf   = (float*)(ws + 0);        // 1600 f
    int*   inv_out = (int*)  (ws + 6400);     // 3200 i
    float* ysum    = (float*)(ws + 19200);    // 3200 f  -- zeroed
    float* ysumsq  = (float*)(ws + 32000);    // 3200 f  -- zeroed
    float* S1      = (float*)(ws + 44800);    // 64 f    -- zeroed
    float* S2      = (float*)(ws + 45056);    // 4096 f  -- zeroed
    float* a_arr   = (float*)(ws + 61440);    // 3200 f
    float* b2_arr  = (float*)(ws + 74240);    // 3200 f
    float* ra      = (float*)(ws + 87040);    // 128 f
    float* rb      = (float*)(ws + 87552);    // 128 f

    hipMemsetAsync(ws + 19200, 0, 42240, stream);
    init_kernel<<<1, 256, 0, stream>>>(fmask, shift_out, maskf, inv_out);

    dim3 gA(N_ / (16 * 8), V_);               // (128, 25)
    gemm1_kernel<<<gA, 256, 0, stream>>>(x0, W, lin_b, shift_in, maskf, inv_out,
                                         out, ysum, ysumsq);

    moment_kernel<<<NR_ / 1024, 512, 0, stream>>>(x0, S1, S2);

    finalize_kernel<<<1, 256, 0, stream>>>(ysum, ysumsq, bn_gamma, bn_beta,
                                           S1, S2, down_w, down_b,
                                           dbn_gamma, dbn_beta,
                                           a_arr, b2_arr, ra, rb);

    gemm2_fuse_kernel<<<NR_ / 16, 256, 0, stream>>>(x0, down_w, a_arr, b2_arr,
                                                    ra, rb, out);
}